// DripBlock_48885317763863
// MI455X (gfx1250) — compile-verified
//
#include <hip/hip_runtime.h>
#include <hip/hip_bf16.h>

// ---------------------------------------------------------------------------
// Modulated conv block (StyleGAN demod conv) for MI455X / gfx1250, wave32.
// y = lrelu( conv3x3(x, conv_w*gain*s[b,i]*sigma_inv[b,o]) + sn*noise + bias )
// Conv runs as implicit GEMM on V_WMMA_F32_16X16X4_F32 (fp32 in/out).
// ---------------------------------------------------------------------------

#define B_     16
#define IN_C   512
#define OUT_C  512
#define H_     64
#define W_     64
#define WDIM_  512
#define K_     3

typedef __attribute__((ext_vector_type(2))) float v2f;
typedef __attribute__((ext_vector_type(8))) float v8f;

static constexpr float RSQRT_WDIM = 0.04419417382415922f;   // 1/sqrt(512)
static constexpr float GAIN       = 0.014731391274719739f;  // 1/sqrt(512*9)
static constexpr float GAIN2      = 1.0f / 4608.0f;
static constexpr float EPS_       = 1e-8f;
static constexpr float SLOPE_     = 0.2f;

// --------------------------- style GEMM: s[b,i] -----------------------------
__global__ void style_kernel(const float* __restrict__ w,
                             const float* __restrict__ lw,
                             const float* __restrict__ lb,
                             float* __restrict__ s) {
  int t = blockIdx.x * blockDim.x + threadIdx.x;    // 16*512 threads
  int b = t >> 9, i = t & 511;
  const float* wr = w + (size_t)b * WDIM_;
  const float* lr = lw + (size_t)i * WDIM_;
  float acc = 0.f;
#pragma unroll 4
  for (int j = 0; j < WDIM_; j += 4) {
    acc += wr[j] * lr[j] + wr[j + 1] * lr[j + 1] +
           wr[j + 2] * lr[j + 2] + wr[j + 3] * lr[j + 3];
  }
  s[t] = acc * RSQRT_WDIM + lb[i];
}

// ------------------- q[o,i] = sum_{kh,kw} conv_w^2 --------------------------
__global__ void wsq_kernel(const float* __restrict__ conv_w,
                           float* __restrict__ q) {
  int t = blockIdx.x * blockDim.x + threadIdx.x;    // 512*512 threads
  const float* p = conv_w + (size_t)t * 9;
  float acc = 0.f;
#pragma unroll
  for (int j = 0; j < 9; ++j) acc += p[j] * p[j];
  q[t] = acc;
}

// ---------- sigma_inv[b,o] = rsqrt(gain^2 * sum_i q[o,i]*s[b,i]^2 + eps) ----
__global__ void demod_kernel(const float* __restrict__ q,
                             const float* __restrict__ s,
                             float* __restrict__ sig) {
  int t = blockIdx.x * blockDim.x + threadIdx.x;    // 16*512 threads
  int b = t >> 9, o = t & 511;
  const float* qr = q + (size_t)o * IN_C;
  const float* sr = s + (size_t)b * IN_C;
  float acc = 0.f;
#pragma unroll 4
  for (int i = 0; i < IN_C; ++i) {
    float sv = sr[i];
    acc += qr[i] * sv * sv;
  }
  sig[t] = rsqrtf(acc * GAIN2 + EPS_);
}

// --------------------------- main conv kernel -------------------------------
// grid = (H, OUT_C/64, B), block = 256 (8 wave32)
// Each block: Y[b, o0:o0+64, h, 0:64] via implicit GEMM, WMMA f32 16x16x4.
#define MT     64                 // out-channel tile (M)
#define CT     16                 // in-channel chunk
#define WPITCH 148                // 144 k-values per chunk, padded row
#define XPITCH 68                 // 66 padded cols (halo), padded to 68
#define CP     18                 // channel-minor pitch (16 ch + 2 pad)
#define XPOS   (3 * XPITCH)       // 204 (row,col) positions per chunk

__global__ __launch_bounds__(256, 2)
void conv_kernel(const float* __restrict__ x,
                 const float* __restrict__ conv_w,
                 const float* __restrict__ s,
                 const float* __restrict__ sig,
                 const float* __restrict__ noise,
                 const float* __restrict__ scale_noise,
                 const float* __restrict__ bias,
                 float* __restrict__ out) {
  __shared__ float Wlds[MT * WPITCH];       // [o_local][k = pos*16 + ci]
  __shared__ float Xlds[XPOS * CP];         // [(row*68+col)][ci]  channel-minor

  const int h   = blockIdx.x;
  const int o0  = blockIdx.y * MT;
  const int b   = blockIdx.z;
  const int tid = threadIdx.x;
  const int lane = tid & 31;
  const int wv   = tid >> 5;                // wave id 0..7
  const int m_idx = wv >> 1;                // 0..3  (M subtile)
  const int n0    = (wv & 1) * 32;          // two N subtiles: n0, n0+16

  const int lhalf = lane >> 4;              // 0 lower / 1 upper lane half
  const int l15   = lane & 15;

  v8f acc0 = {};
  v8f acc1 = {};

  const float* sr   = s   + (size_t)b * IN_C;
  const float* sigr = sig + (size_t)b * OUT_C;

  // sigma_inv is chunk-invariant: preload per weight-staging row
  const int ci_stage = tid & 15;            // staging channel (fixed)
  float sig4[4];
#pragma unroll
  for (int r = 0; r < 4; ++r) {
    int ol = (tid + (r << 8)) >> 4;
    sig4[r] = GAIN * sigr[o0 + ol];
  }

  const int xci = tid >> 4;                 // staging channel for X (fixed)
  const int xq  = tid & 15;                 // staging position phase

  for (int ci0 = 0; ci0 < IN_C; ci0 += CT) {
    // ---- stage modulated/demodulated weights into LDS ----
    const float sv = sr[ci0 + ci_stage];
#pragma unroll
    for (int r = 0; r < 4; ++r) {
      int ol = (tid + (r << 8)) >> 4;
      float scale = sig4[r] * sv;
      const float* g = conv_w + ((size_t)(o0 + ol) * IN_C + ci0 + ci_stage) * 9;
      float* wd = &Wlds[ol * WPITCH + ci_stage];
#pragma unroll
      for (int pos = 0; pos < 9; ++pos)
        wd[pos * 16] = g[pos] * scale;
    }

    // ---- stage haloed x slab, channel-minor, zero-padded SAME ----
    {
      const float* xc = x + ((size_t)b * IN_C + ci0 + xci) * (H_ * W_);
#pragma unroll
      for (int k = 0; k < 13; ++k) {
        int p = xq + (k << 4);
        if (p < XPOS) {
          int r = p / XPITCH;
          int c = p - r * XPITCH;
          int hh = h + r - 1;
          float v = 0.f;
          if (c >= 1 && c <= 64 && (unsigned)hh < (unsigned)H_)
            v = xc[hh * W_ + (c - 1)];
          Xlds[p * CP + xci] = v;
        }
      }
    }

    // ---- prefetch next chunk (global_prefetch_b8) ----
    if (ci0 + CT < IN_C) {
      const float* nx = x + ((size_t)b * IN_C + ci0 + CT + xci) * (H_ * W_) +
                        (size_t)h * W_ + (xq << 2);
      __builtin_prefetch(nx, 0, 1);
      const float* nw = conv_w +
          ((size_t)(o0 + (tid >> 2)) * IN_C + ci0 + CT + ((tid & 3) << 2)) * 9;
      __builtin_prefetch(nw, 0, 1);
    }
    __syncthreads();

    // ---- WMMA over 9 taps x 4 k-steps of 4 ----
    const float* arow = &Wlds[(m_idx * 16 + l15) * WPITCH + (lhalf << 1)];
    const float* xcol = &Xlds[(n0 + l15) * CP + (lhalf << 1)];
#pragma unroll
    for (int pos = 0; pos < 9; ++pos) {
      const int kh = pos / 3;
      const int kw = pos - kh * 3;
      const float* bbase = xcol + (kh * XPITCH + kw) * CP;
#pragma unroll
      for (int c4 = 0; c4 < 16; c4 += 4) {
        // A: lanes 0-15 hold K=c4+0/1, lanes 16-31 K=c4+2/3 (8B-aligned b64)
        v2f a = *(const v2f*)(arow + pos * 16 + c4);
        // B: VGPR0 = K rows {0 lower, 2 upper}, VGPR1 = {1, 3}; contiguous b64
        v2f b0 = *(const v2f*)(bbase + c4);
        v2f b1 = *(const v2f*)(bbase + 16 * CP + c4);
        acc0 = __builtin_amdgcn_wmma_f32_16x16x4_f32(
            false, a, false, b0, (short)0, acc0, false, false);
        acc1 = __builtin_amdgcn_wmma_f32_16x16x4_f32(
            false, a, false, b1, (short)0, acc1, false, false);
      }
    }
    __syncthreads();
  }

  // ---- epilogue: noise, bias, leaky relu, store ----
  const float nz0 = noise[((size_t)b * H_ + h) * W_ + n0 + l15];
  const float nz1 = noise[((size_t)b * H_ + h) * W_ + n0 + 16 + l15];
  const int obase = o0 + m_idx * 16 + lhalf * 8;
#pragma unroll
  for (int r = 0; r < 8; ++r) {
    int o = obase + r;                       // C layout: VGPR r -> M=r (+8 upper)
    float sn = scale_noise[o];
    float bs = bias[o];
    float v0 = acc0[r] + sn * nz0 + bs;
    float v1 = acc1[r] + sn * nz1 + bs;
    v0 = v0 > 0.f ? v0 : v0 * SLOPE_;
    v1 = v1 > 0.f ? v1 : v1 * SLOPE_;
    size_t base = (((size_t)b * OUT_C + o) * H_ + h) * W_;
    out[base + n0 + l15]      = v0;
    out[base + n0 + 16 + l15] = v1;
  }
}

// ---------------------------------------------------------------------------
extern "C" void kernel_launch(void* const* d_in, const int* in_sizes, int n_in,
                              void* d_out, int out_size, void* d_ws, size_t ws_size,
                              hipStream_t stream) {
  const float* x           = (const float*)d_in[0];
  const float* w           = (const float*)d_in[1];
  const float* noise       = (const float*)d_in[2];
  const float* linear_w    = (const float*)d_in[3];
  const float* linear_b    = (const float*)d_in[4];
  const float* conv_w      = (const float*)d_in[5];
  const float* scale_noise = (const float*)d_in[6];
  const float* bias        = (const float*)d_in[7];
  float* out = (float*)d_out;

  float* s_ws   = (float*)d_ws;                 // 16*512
  float* sig_ws = s_ws + B_ * IN_C;             // 16*512
  float* q_ws   = sig_ws + B_ * OUT_C;          // 512*512

  style_kernel<<<(B_ * IN_C) / 256, 256, 0, stream>>>(w, linear_w, linear_b, s_ws);
  wsq_kernel<<<(OUT_C * IN_C) / 256, 256, 0, stream>>>(conv_w, q_ws);
  demod_kernel<<<(B_ * OUT_C) / 256, 256, 0, stream>>>(q_ws, s_ws, sig_ws);

  dim3 grid(H_, OUT_C / MT, B_);                // 64 x 8 x 16
  conv_kernel<<<grid, 256, 0, stream>>>(x, conv_w, s_ws, sig_ws, noise,
                                        scale_noise, bias, out);
}